// RGCN_6382321402260
// MI455X (gfx1250) — compile-verified
//
#include <hip/hip_runtime.h>
#include <math.h>

#define N_NODES   100000
#define N_EDGES   1200000
#define DIM       64
#define NREL      4
#define NGRAPH    512

typedef __attribute__((ext_vector_type(2))) float v2f;
typedef __attribute__((ext_vector_type(8))) float v8f;

// ---------------------------------------------------------------------------
// Zero a float buffer (grid-stride)
// ---------------------------------------------------------------------------
__global__ void zero_kernel(float* __restrict__ p, long long n) {
    long long i = (long long)blockIdx.x * blockDim.x + threadIdx.x;
    long long stride = (long long)gridDim.x * blockDim.x;
    for (; i < n; i += stride) p[i] = 0.0f;
}

// ---------------------------------------------------------------------------
// h0[n, d] = emb[atom_type[n], d]
// ---------------------------------------------------------------------------
__global__ void embed_kernel(const int* __restrict__ atom_type,
                             const float* __restrict__ emb,
                             float* __restrict__ h0) {
    long long idx = (long long)blockIdx.x * blockDim.x + threadIdx.x;
    if (idx >= (long long)N_NODES * DIM) return;
    int n = (int)(idx >> 6);
    int d = (int)(idx & 63);
    h0[idx] = emb[(long long)atom_type[n] * DIM + d];
}

// ---------------------------------------------------------------------------
// Edge scatter: one wave32 per edge.  Lane l carries features [2l, 2l+1].
// sums[(dst*R + etype), :] += x[src, :]  ;  cnt[dst*R + etype] += 1
// x (25.6 MB) and sums (102.4 MB) are L2-resident (192 MB L2): this phase is
// L2-atomic bound, the HBM side only streams 14.4 MB of edge indices.
// ---------------------------------------------------------------------------
__global__ __launch_bounds__(256)
void edge_scatter_kernel(const float* __restrict__ x,
                         const int* __restrict__ src,
                         const int* __restrict__ dst,
                         const int* __restrict__ etype,
                         float* __restrict__ sums,
                         float* __restrict__ cnt) {
    int wave = (int)(((long long)blockIdx.x * blockDim.x + threadIdx.x) >> 5);
    int lane = threadIdx.x & 31;
    if (wave >= N_EDGES) return;

    int s = src[wave];
    int d = dst[wave];
    int r = etype[wave];

    const float2 v = ((const float2*)(x + (size_t)s * DIM))[lane];
    float* base = sums + ((size_t)d * NREL + r) * DIM + 2 * lane;
    atomicAdd(base + 0, v.x);
    atomicAdd(base + 1, v.y);
    if (lane == 0) atomicAdd(cnt + (size_t)d * NREL + r, 1.0f);
}

// ---------------------------------------------------------------------------
// Transform: one wave32 per 16-node tile.
//   out[16,64] = sum_r mean_r[16,64] @ Wrel[r] + x[16,64] @ Wroot + bias
// via V_WMMA_F32_16X16X4_F32 (bit-exact fp32; compute is trivial vs the
// scatter phase so there is no reason to drop precision).
// All 5 weight matrices (80 KB) are staged into LDS once per workgroup
// (8 waves share), so B fragments come from ds_load instead of per-wave
// global traffic.  A 16x4 frag: lane(0..15)->M=lane, K={k,k+1};
// lane(16..31)->M=lane-16, K={k+2,k+3}.  B 4x16 symmetric by column.
// C/D: 8 VGPRs, M = vgpr + 8*(lane>=16).
// ---------------------------------------------------------------------------
__global__ __launch_bounds__(256)
void transform_kernel(const float* __restrict__ x,
                      const float* __restrict__ sums,
                      const float* __restrict__ cnt,
                      const float* __restrict__ wrel,   // [R, D, D] (i,o row-major)
                      const float* __restrict__ wroot,  // [D, D]
                      const float* __restrict__ bias,   // [D]
                      float* __restrict__ out,
                      int apply_tanh) {
    __shared__ float sW[(NREL + 1) * DIM * DIM];        // 80 KB of 320 KB WGP LDS

    // cooperative weight staging (before any wave exits)
    {
        float4* s4 = (float4*)sW;
        const float4* wr4 = (const float4*)wrel;
        const float4* wo4 = (const float4*)wroot;
        const int nrel4 = NREL * DIM * DIM / 4;         // 4096 float4
        const int nroot4 = DIM * DIM / 4;               // 1024 float4
        for (int i = threadIdx.x; i < nrel4; i += blockDim.x)  s4[i] = wr4[i];
        for (int i = threadIdx.x; i < nroot4; i += blockDim.x) s4[nrel4 + i] = wo4[i];
    }
    __syncthreads();

    const int tile = blockIdx.x * (blockDim.x >> 5) + (threadIdx.x >> 5);
    const int lane = threadIdx.x & 31;
    if (tile * 16 >= N_NODES) return;           // uniform per wave: EXEC stays full

    const int m     = lane & 15;                // A row / D column index
    const int col   = lane & 15;
    const int khalf = (lane >> 4) << 1;         // 0 or 2
    const int node  = tile * 16 + m;

    // accumulators for the 4 output column tiles, seeded with bias
    v8f c0, c1, c2, c3;
    {
        float b0v = bias[col +  0], b1v = bias[col + 16];
        float b2v = bias[col + 32], b3v = bias[col + 48];
        for (int j = 0; j < 8; ++j) { c0[j] = b0v; c1[j] = b1v; c2[j] = b2v; c3[j] = b3v; }
    }

    // per-relation mean scales (counts are contiguous per node), root scale = 1
    float sc[NREL + 1];
    {
        float4 cv = ((const float4*)cnt)[node];
        sc[0] = 1.0f / fmaxf(cv.x, 1.0f);
        sc[1] = 1.0f / fmaxf(cv.y, 1.0f);
        sc[2] = 1.0f / fmaxf(cv.z, 1.0f);
        sc[3] = 1.0f / fmaxf(cv.w, 1.0f);
        sc[4] = 1.0f;
    }

    for (int rr = 0; rr < NREL + 1; ++rr) {     // 4 relations + root
        const float* Asrc = (rr < NREL) ? (sums + ((size_t)node * NREL + rr) * DIM)
                                        : (x + (size_t)node * DIM);
        const float scale = sc[rr];
        const float* W = sW + rr * DIM * DIM;   // LDS

        for (int k4 = 0; k4 < DIM; k4 += 4) {
            const int kk = k4 + khalf;
            v2f a;
            a[0] = Asrc[kk]     * scale;
            a[1] = Asrc[kk + 1] * scale;
            const float* w0 = W + kk * DIM;     // row K=kk   (LDS)
            const float* w1 = w0 + DIM;         // row K=kk+1 (LDS)
            v2f b;
            b[0] = w0[col +  0]; b[1] = w1[col +  0];
            c0 = __builtin_amdgcn_wmma_f32_16x16x4_f32(false, a, false, b, (short)0, c0, false, false);
            b[0] = w0[col + 16]; b[1] = w1[col + 16];
            c1 = __builtin_amdgcn_wmma_f32_16x16x4_f32(false, a, false, b, (short)0, c1, false, false);
            b[0] = w0[col + 32]; b[1] = w1[col + 32];
            c2 = __builtin_amdgcn_wmma_f32_16x16x4_f32(false, a, false, b, (short)0, c2, false, false);
            b[0] = w0[col + 48]; b[1] = w1[col + 48];
            c3 = __builtin_amdgcn_wmma_f32_16x16x4_f32(false, a, false, b, (short)0, c3, false, false);
        }
    }

    // write-out (optionally tanh for layer 0)
    const int rbase = tile * 16 + ((lane >> 4) << 3);
    for (int j = 0; j < 8; ++j) {
        float v0 = c0[j], v1 = c1[j], v2 = c2[j], v3 = c3[j];
        if (apply_tanh) {
            v0 = tanhf(v0); v1 = tanhf(v1); v2 = tanhf(v2); v3 = tanhf(v3);
        }
        float* orow = out + (size_t)(rbase + j) * DIM + col;
        orow[ 0] = v0;
        orow[16] = v1;
        orow[32] = v2;
        orow[48] = v3;
    }
}

// ---------------------------------------------------------------------------
// Pooling: one wave32 per node; feature mean via wave shuffle reduction,
// scatter-add into per-graph sum / count.
// ---------------------------------------------------------------------------
__global__ __launch_bounds__(256)
void pool_nodes_kernel(const float* __restrict__ h,
                       const int* __restrict__ batch,
                       float* __restrict__ psum,
                       float* __restrict__ pcnt) {
    int node = (int)(((long long)blockIdx.x * blockDim.x + threadIdx.x) >> 5);
    int lane = threadIdx.x & 31;
    if (node >= N_NODES) return;
    float2 v = ((const float2*)(h + (size_t)node * DIM))[lane];
    float t = v.x + v.y;
    for (int off = 16; off > 0; off >>= 1) t += __shfl_xor(t, off, 32);
    if (lane == 0) {
        int g = batch[node];
        atomicAdd(psum + g, t * (1.0f / (float)DIM));
        atomicAdd(pcnt + g, 1.0f);
    }
}

__global__ void finalize_kernel(const float* __restrict__ psum,
                                const float* __restrict__ pcnt,
                                float* __restrict__ out) {
    int g = blockIdx.x * blockDim.x + threadIdx.x;
    if (g < NGRAPH) out[g] = psum[g] / fmaxf(pcnt[g], 1.0f);
}

// ---------------------------------------------------------------------------
// Launch
// ---------------------------------------------------------------------------
extern "C" void kernel_launch(void* const* d_in, const int* in_sizes, int n_in,
                              void* d_out, int out_size, void* d_ws, size_t ws_size,
                              hipStream_t stream) {
    const int*   atom_type = (const int*)  d_in[0];
    const int*   edge_idx  = (const int*)  d_in[1];   // [2, E]
    const int*   edge_type = (const int*)  d_in[2];
    const int*   batch     = (const int*)  d_in[3];
    const float* emb       = (const float*)d_in[4];
    const float* wrel0     = (const float*)d_in[5];
    const float* wroot0    = (const float*)d_in[6];
    const float* b0        = (const float*)d_in[7];
    const float* wrel1     = (const float*)d_in[8];
    const float* wroot1    = (const float*)d_in[9];
    const float* b1        = (const float*)d_in[10];
    float* out = (float*)d_out;

    const int* src = edge_idx;
    const int* dst = edge_idx + N_EDGES;

    // workspace layout (floats); total ~155 MB
    float* ws   = (float*)d_ws;
    float* h0   = ws;                                   // [N, D]
    float* h1   = h0   + (size_t)N_NODES * DIM;         // [N, D]
    float* sums = h1   + (size_t)N_NODES * DIM;         // [N, R, D]
    float* cnt  = sums + (size_t)N_NODES * NREL * DIM;  // [N, R]
    float* psum = cnt  + (size_t)N_NODES * NREL;        // [G]
    float* pcnt = psum + NGRAPH;                        // [G]

    const int TB = 256;
    const int edgeBlocks  = (N_EDGES * 32 + TB - 1) / TB;
    const int tileBlocks  = ((N_NODES / 16) + (TB / 32) - 1) / (TB / 32);
    const int nodeWBlocks = (N_NODES * 32 + TB - 1) / TB;
    const long long sumsN = (long long)N_NODES * NREL * DIM + (long long)N_NODES * NREL;

    // embed
    embed_kernel<<<(N_NODES * DIM + TB - 1) / TB, TB, 0, stream>>>(atom_type, emb, h0);

    // ---- layer 0 ----
    zero_kernel<<<4096, TB, 0, stream>>>(sums, sumsN);  // sums + cnt contiguous
    edge_scatter_kernel<<<edgeBlocks, TB, 0, stream>>>(h0, src, dst, edge_type, sums, cnt);
    transform_kernel<<<tileBlocks, TB, 0, stream>>>(h0, sums, cnt, wrel0, wroot0, b0, h1, 1);

    // ---- layer 1 ----
    zero_kernel<<<4096, TB, 0, stream>>>(sums, sumsN);
    edge_scatter_kernel<<<edgeBlocks, TB, 0, stream>>>(h1, src, dst, edge_type, sums, cnt);
    transform_kernel<<<tileBlocks, TB, 0, stream>>>(h1, sums, cnt, wrel1, wroot1, b1, h0, 0);

    // ---- pooling ----
    zero_kernel<<<2, TB, 0, stream>>>(psum, 2 * NGRAPH);
    pool_nodes_kernel<<<nodeWBlocks, TB, 0, stream>>>(h0, batch, psum, pcnt);
    finalize_kernel<<<(NGRAPH + TB - 1) / TB, TB, 0, stream>>>(psum, pcnt, out);
}